// CustomGridSampleModel_11450382812072
// MI455X (gfx1250) — compile-verified
//
#include <hip/hip_runtime.h>
#include <hip/hip_bf16.h>
#include <stdint.h>

// Problem constants (from reference): x[N,C,H,W], grid[N,Ho,Wo,2], out[N,C,Ho,Wo]
#define N_  8
#define C_  64
#define H_  256
#define W_  256
#define HW_ (H_ * W_)          // 65536 == Ho*Wo as well

// 8-byte vector with only 4-byte alignment guarantee: lets the backend use
// global_load_b64 (gfx1250 supports unaligned VMEM) while staying legal C++.
typedef float f2a4 __attribute__((ext_vector_type(2), aligned(4)));

__global__ __launch_bounds__(256)
void grid_sample_nchw(const float* __restrict__ x,
                      const float* __restrict__ grid,
                      float* __restrict__ out)
{
    __shared__ float sgrid[512];                 // one (n,ho) row of grid: 256 * float2

    const int tid = threadIdx.x;
    const int pix = blockIdx.x * 256 + tid;      // pix = (n*Ho + ho)*Wo + wo
    const int n   = pix >> 16;                   // Ho*Wo == 65536
    const int sp  = pix & 0xFFFF;                // ho*Wo + wo

    // ---- Stage this block's 2KB grid row into LDS with the CDNA5 async
    //      global->LDS DMA path (ASYNCcnt), then wait on our own transfer.
    {
        uint64_t gaddr = (uint64_t)(uintptr_t)(grid + (size_t)pix * 2);
        uint32_t loff  = (uint32_t)(uintptr_t)(&sgrid[2 * tid]);  // low 32 bits = LDS byte offset
        asm volatile(
            "global_load_async_to_lds_b64 %0, %1, off\n\t"
            "s_wait_asynccnt 0"
            :: "v"(loff), "v"(gaddr)
            : "memory");
    }
    const float gx = sgrid[2 * tid + 0];
    const float gy = sgrid[2 * tid + 1];

    // ---- Unnormalize (align_corners=True) and build bilinear weights once.
    const float ix  = (gx + 1.0f) * (0.5f * (W_ - 1));
    const float iy  = (gy + 1.0f) * (0.5f * (H_ - 1));
    const float x0f = floorf(ix);
    const float y0f = floorf(iy);
    const float wx1 = ix - x0f, wy1 = iy - y0f;
    const float wx0 = 1.0f - wx1, wy0 = 1.0f - wy1;

    int x0 = (int)x0f; x0 = x0 < 0 ? 0 : (x0 > W_ - 1 ? W_ - 1 : x0);
    int y0 = (int)y0f; y0 = y0 < 0 ? 0 : (y0 > H_ - 1 ? H_ - 1 : y0);
    const int y1 = (y0 + 1 > H_ - 1) ? H_ - 1 : y0 + 1;

    // Pair-load trick: read (y, xb) and (y, xb+1) with ONE b64 load.
    // At the right border (x0 == W-1) wx1 is exactly 0, so fold the clamp into
    // the weights (use pair.y with the full wy weight) and shift the load to
    // column W-2 so it never reads out of bounds. Inner loop stays branch-free.
    const int  xb = (x0 > W_ - 2) ? W_ - 2 : x0;
    const bool hi = (x0 == W_ - 1);
    const float w00 = hi ? 0.0f : wy0 * wx0;   // weight of pair0.x
    const float w01 = hi ? wy0  : wy0 * wx1;   // weight of pair0.y
    const float w10 = hi ? 0.0f : wy1 * wx0;   // weight of pair1.x
    const float w11 = hi ? wy1  : wy1 * wx1;   // weight of pair1.y

    const unsigned o0 = (unsigned)(y0 * W_ + xb);
    const unsigned o1 = (unsigned)(y1 * W_ + xb);

    const float* __restrict__ px = x   + (size_t)n * (C_ * HW_);
    float*       __restrict__ po = out + (size_t)n * (C_ * HW_) + sp;

    // One-shot prefetch (global_prefetch_b8) of a couple of mid-loop channels
    // to warm L2 on the cold pass; negligible issue cost.
    __builtin_prefetch(px +  8 * HW_ + o0, 0, 1);
    __builtin_prefetch(px +  8 * HW_ + o1, 0, 1);
    __builtin_prefetch(px + 32 * HW_ + o0, 0, 1);
    __builtin_prefetch(px + 32 * HW_ + o1, 0, 1);

    // ---- 64 channels: 2 gathered b64 loads + 4-term FMA + 1 NT store each.
    // Non-temporal stores keep the 128MiB output stream from evicting the
    // 128MiB input x, which fits (with the grid) in the 192MB L2.
#pragma unroll 4
    for (int c = 0; c < C_; ++c) {
        const float* pc = px + (size_t)c * HW_;
        const f2a4 r0 = *(const f2a4*)(pc + o0);   // row y0: cols xb, xb+1
        const f2a4 r1 = *(const f2a4*)(pc + o1);   // row y1: cols xb, xb+1
        const float v = r0.x * w00 + r0.y * w01 + r1.x * w10 + r1.y * w11;
        __builtin_nontemporal_store(v, po + (size_t)c * HW_);
    }
}

extern "C" void kernel_launch(void* const* d_in, const int* in_sizes, int n_in,
                              void* d_out, int out_size, void* d_ws, size_t ws_size,
                              hipStream_t stream) {
    const float* x    = (const float*)d_in[0];   // [8,64,256,256] f32
    const float* grid = (const float*)d_in[1];   // [8,256,256,2]  f32
    float*       out  = (float*)d_out;           // [8,64,256,256] f32

    const int pixels = N_ * H_ * W_;             // 524288 output pixels
    grid_sample_nchw<<<dim3(pixels / 256), dim3(256), 0, stream>>>(x, grid, out);
}